// LSTMModel_1228360647128
// MI455X (gfx1250) — compile-verified
//
#include <hip/hip_runtime.h>
#include <math.h>

// CDNA5 / gfx1250 persistent LSTM.
// - W_hh (16 MB) distributed across 64 WGPs, resident in LDS (256 KB/WGP slice).
// - Recurrent GEMV done with V_WMMA_F32_16X16X4_F32 (full fp32 precision).
// - 8192 grid barriers via agent-scope atomic counter (persistent kernel).

typedef __attribute__((ext_vector_type(2))) float v2f;
typedef __attribute__((ext_vector_type(8))) float v8f;

#define T_STEPS   8192
#define HID       1024
#define NWG       64          // workgroups; each owns 16 hidden units
#define UNITS     16
#define NTHREADS  256         // 8 wave32s: 4 gates x 2 K-halves

// Dynamic LDS layout (in floats):
//   sW    : 4 gates * 256 k-tiles * 32 lanes * 2  = 65536   (256 KB, WMMA-A swizzled)
//   sh    : 1024                                            (h_{t-1} broadcast)
//   swfc  : 1024                                            (W_fc, used by WG0)
//   sbsum : 64   (b_ih + b_hh for this WG's 64 gate rows)
//   swih  : 64   (W_ih column, IN==1)
//   sc    : 16   (cell state, private to WG, lives across steps)
//   sgp   : 8*16 (per-wave gate partials, column 0 of C tiles)
//   swred : 8    (WG0 dot-product wave partials)
#define LDS_FLOATS (65536 + 1024 + 1024 + 64 + 64 + 16 + 128 + 8)

extern "C" __global__ void lstm1250_init(float* hbuf, unsigned* cnt) {
    int i = blockIdx.x * blockDim.x + threadIdx.x;
    if (i < 2 * HID) hbuf[i] = 0.0f;
    if (i == 0) *cnt = 0u;
}

extern "C" __global__ __launch_bounds__(NTHREADS)
void lstm1250_persistent(const float* __restrict__ x,
                         const float* __restrict__ W_ih,
                         const float* __restrict__ W_hh,
                         const float* __restrict__ b_ih,
                         const float* __restrict__ b_hh,
                         const float* __restrict__ W_fc,
                         const float* __restrict__ b_fc,
                         float* __restrict__ out,
                         float* __restrict__ hbuf,   // 2*HID, double buffered
                         unsigned* __restrict__ cnt) // grid barrier counter
{
    extern __shared__ float smem[];
    float* sW    = smem;
    float* sh    = smem + 65536;
    float* swfc  = sh + HID;
    float* sbsum = swfc + HID;
    float* swih  = sbsum + 64;
    float* sc    = swih + 64;
    float* sgp   = sc + 16;      // [8][16]
    float* swred = sgp + 128;    // [8]

    const int tid   = threadIdx.x;
    const int wg    = blockIdx.x;
    const int base  = wg * UNITS;            // first hidden unit owned
    const int lane  = tid & 31;
    const int w     = tid >> 5;              // wave 0..7
    const int gate  = w & 3;
    const int khalf = w >> 2;
    const int bsel  = (lane >= 16) ? 2 : 0;  // B operand K-pair select

    // ---- stage W_hh slice into LDS in WMMA-A tile order --------------------
    // Tile (gate, kt) covers rows base..base+15, K = 4*kt..4*kt+3.
    // A layout (f32 16x16x4): lanes 0-15 -> M=lane, VGPR0/1 = K0/K1;
    //                         lanes 16-31 -> M=lane-16, VGPR0/1 = K2/K3.
    for (int e = tid; e < 4 * UNITS * HID; e += NTHREADS) {
        int g   = e >> 14;           // 16384 elems per gate
        int rem = e & 16383;
        int m   = rem >> 10;
        int k   = rem & 1023;
        float v = W_hh[(size_t)(g * HID + base + m) * HID + k];
        int kt = k >> 2, kj = k & 3;
        int ln = ((kj >> 1) << 4) + m;   // 0..31
        int j  = kj & 1;
        sW[(size_t)(((g << 8) + kt) * 32 + ln) * 2 + j] = v;
    }
    for (int i = tid; i < HID; i += NTHREADS) swfc[i] = W_fc[i];
    if (tid < 64) {
        int g = tid >> 4, m = tid & 15;
        int row = g * HID + base + m;
        sbsum[tid] = b_ih[row] + b_hh[row];
        swih[tid]  = W_ih[row];          // IN == 1
    }
    if (tid < 16) sc[tid] = 0.0f;
    __syncthreads();

    const float bfc = b_fc[0];

    for (int t = 0; t < T_STEPS; ++t) {
        const int rbuf = t & 1;
        // ---- broadcast-read h_{t-1} (DEV scope -> bypass stale WGP$) -------
        for (int i = tid; i < HID; i += NTHREADS)
            sh[i] = __hip_atomic_load(&hbuf[rbuf * HID + i],
                                      __ATOMIC_RELAXED, __HIP_MEMORY_SCOPE_AGENT);
        __syncthreads();

        // ---- recurrent GEMV via fp32 WMMA ---------------------------------
        v8f a0 = {}, a1 = {}, a2 = {}, a3 = {};
        const float* wbase = sW + (size_t)(((gate << 8) + (khalf << 7))) * 64;
        #pragma unroll 2
        for (int kt = 0; kt < 128; kt += 4) {
            const float* p = wbase + (size_t)kt * 64;
            v2f A0 = *(const v2f*)(p +        lane * 2);
            v2f A1 = *(const v2f*)(p +  64 +  lane * 2);
            v2f A2 = *(const v2f*)(p + 128 +  lane * 2);
            v2f A3 = *(const v2f*)(p + 192 +  lane * 2);
            int k0 = ((khalf << 7) + kt) << 2;
            v2f B0 = *(const v2f*)(sh + k0      + bsel);  // broadcast reads
            v2f B1 = *(const v2f*)(sh + k0 + 4  + bsel);
            v2f B2 = *(const v2f*)(sh + k0 + 8  + bsel);
            v2f B3 = *(const v2f*)(sh + k0 + 12 + bsel);
            a0 = __builtin_amdgcn_wmma_f32_16x16x4_f32(false, A0, false, B0, (short)0, a0, false, false);
            a1 = __builtin_amdgcn_wmma_f32_16x16x4_f32(false, A1, false, B1, (short)0, a1, false, false);
            a2 = __builtin_amdgcn_wmma_f32_16x16x4_f32(false, A2, false, B2, (short)0, a2, false, false);
            a3 = __builtin_amdgcn_wmma_f32_16x16x4_f32(false, A3, false, B3, (short)0, a3, false, false);
        }
        v8f cs = (a0 + a1) + (a2 + a3);

        // Column N=0 of C: lane0 has M=0..7 in v0..v7, lane16 has M=8..15.
        if (lane == 0 || lane == 16) {
            int mb = (lane >> 4) << 3;
            #pragma unroll
            for (int r = 0; r < 8; ++r) sgp[w * 16 + mb + r] = cs[r];
        }

        // WG0: out[t-1] = h_{t-1} . W_fc + b_fc (one step behind)
        float dotp = 0.0f;
        if (wg == 0) {
            for (int i = tid; i < HID; i += NTHREADS) dotp += sh[i] * swfc[i];
            #pragma unroll
            for (int off = 16; off > 0; off >>= 1) dotp += __shfl_xor(dotp, off, 32);
            if (lane == 0) swred[w] = dotp;
        }
        __syncthreads();

        // ---- pointwise LSTM update for the 16 owned units -----------------
        if (tid < 16) {
            int m = tid;
            float xt = x[t];
            float gi = sgp[0 * 16 + m] + sgp[4 * 16 + m] + xt * swih[0  + m] + sbsum[0  + m];
            float gf = sgp[1 * 16 + m] + sgp[5 * 16 + m] + xt * swih[16 + m] + sbsum[16 + m];
            float gg = sgp[2 * 16 + m] + sgp[6 * 16 + m] + xt * swih[32 + m] + sbsum[32 + m];
            float go = sgp[3 * 16 + m] + sgp[7 * 16 + m] + xt * swih[48 + m] + sbsum[48 + m];
            float si = 1.0f / (1.0f + expf(-gi));
            float sf = 1.0f / (1.0f + expf(-gf));
            float so = 1.0f / (1.0f + expf(-go));
            float cc = sf * sc[m] + si * tanhf(gg);
            sc[m] = cc;
            float hh = so * tanhf(cc);
            __hip_atomic_store(&hbuf[(rbuf ^ 1) * HID + base + m], hh,
                               __ATOMIC_RELAXED, __HIP_MEMORY_SCOPE_AGENT);
        } else if (wg == 0 && tid == 16 && t > 0) {
            float s = bfc;
            #pragma unroll
            for (int i = 0; i < 8; ++i) s += swred[i];
            out[t - 1] = s;
        }

        // ---- grid barrier (release h, then arrive/spin) -------------------
        __threadfence();
        __syncthreads();
        if (tid == 0) {
            __hip_atomic_fetch_add(cnt, 1u, __ATOMIC_ACQ_REL, __HIP_MEMORY_SCOPE_AGENT);
            unsigned target = (unsigned)(t + 1) * NWG;
            while (__hip_atomic_load(cnt, __ATOMIC_ACQUIRE, __HIP_MEMORY_SCOPE_AGENT) < target)
                __builtin_amdgcn_s_sleep(2);
        }
        __syncthreads();
    }

    // ---- tail: out[T-1] from final h (written to buffer 0 at t=8191) ------
    if (wg == 0) {
        float dotp = 0.0f;
        for (int i = tid; i < HID; i += NTHREADS)
            dotp += __hip_atomic_load(&hbuf[i], __ATOMIC_RELAXED,
                                      __HIP_MEMORY_SCOPE_AGENT) * swfc[i];
        #pragma unroll
        for (int off = 16; off > 0; off >>= 1) dotp += __shfl_xor(dotp, off, 32);
        if (lane == 0) swred[w] = dotp;
        __syncthreads();
        if (tid == 0) {
            float s = bfc;
            #pragma unroll
            for (int i = 0; i < 8; ++i) s += swred[i];
            out[T_STEPS - 1] = s;
        }
    }
}

extern "C" void kernel_launch(void* const* d_in, const int* in_sizes, int n_in,
                              void* d_out, int out_size, void* d_ws, size_t ws_size,
                              hipStream_t stream) {
    const float* x    = (const float*)d_in[0];
    const float* Wih  = (const float*)d_in[1];
    const float* Whh  = (const float*)d_in[2];
    const float* bih  = (const float*)d_in[3];
    const float* bhh  = (const float*)d_in[4];
    const float* Wfc  = (const float*)d_in[5];
    const float* bfc  = (const float*)d_in[6];
    float* out = (float*)d_out;

    float*    hbuf = (float*)d_ws;                                // 2*HID floats
    unsigned* cnt  = (unsigned*)((char*)d_ws + 2 * HID * sizeof(float));

    lstm1250_init<<<(2 * HID + 255) / 256, 256, 0, stream>>>(hbuf, cnt);

    size_t ldsBytes = (size_t)LDS_FLOATS * sizeof(float);         // ~265 KB < 320 KB
    lstm1250_persistent<<<NWG, NTHREADS, ldsBytes, stream>>>(
        x, Wih, Whh, bih, bhh, Wfc, bfc, out, hbuf, cnt);
}